// RRPlus_M34res_48515950576140
// MI455X (gfx1250) — compile-verified
//
#include <hip/hip_runtime.h>

// ---------------------------------------------------------------------------
// MI455X (gfx1250) implementation of the scale-equivariant 1D ResNet.
// Conv compute: V_WMMA_F32_16X16X4_F32 (fp32-exact path).
// Weight tiles staged into LDS by the Tensor Data Mover (TENSOR_LOAD_TO_LDS),
// synchronized with s_wait_tensorcnt + workgroup barrier; 4 waves per block
// share one tile and cover 4 adjacent L-tiles.
// ---------------------------------------------------------------------------

typedef float v2f  __attribute__((ext_vector_type(2)));
typedef float v8f  __attribute__((ext_vector_type(8)));
typedef unsigned int v4u __attribute__((ext_vector_type(4)));
typedef int  v4i  __attribute__((ext_vector_type(4)));
typedef int  v8i  __attribute__((ext_vector_type(8)));

#define BN_EPS 2e-05f

// ---------------------------------------------------------------------------
// TDM: stage a [16 x cols] fp32 tile (row stride rowStride elements, rowsAvail
// valid rows -> rows beyond zero-fill per D# OOB rule) from global to LDS.
// D# packing per CDNA5 ISA.
// ---------------------------------------------------------------------------
__device__ __forceinline__ void tdm_load_w_tile(const float* gptr,
                                                unsigned ldsAddr,
                                                int rowsAvail, int cols,
                                                int rowStride) {
#if __has_builtin(__builtin_amdgcn_tensor_load_to_lds)
  const unsigned long long ga = (unsigned long long)(size_t)gptr;
  v4u g0;
  g0[0] = 1u;                                    // count=1 (valid), user mode
  g0[1] = ldsAddr;                               // lds_addr [63:32]
  g0[2] = (unsigned)(ga & 0xffffffffull);        // global_addr [95:64]
  g0[3] = (unsigned)((ga >> 32) & 0x1ffffffull)  // global_addr [120:96]
        | (2u << 30);                            // type=2 ("image")
  const unsigned ucols = (unsigned)cols;
  const unsigned urows = (unsigned)rowsAvail;
  const unsigned ustr  = (unsigned)rowStride;
  v8i g1;
  g1[0] = (int)(2u << 16);                       // wg_mask=0, data_size=2 (4B)
  g1[1] = (int)((ucols & 0xffffu) << 16);        // tensor_dim0 [15:0]
  g1[2] = (int)(((ucols >> 16) & 0xffffu)        // tensor_dim0 [31:16]
        | ((urows & 0xffffu) << 16));            // tensor_dim1 [15:0]
  g1[3] = (int)(((urows >> 16) & 0xffffu)        // tensor_dim1 [31:16]
        | ((ucols & 0xffffu) << 16));            // tile_dim0 = cols
  g1[4] = (int)16u;                              // tile_dim1=16, tile_dim2=0
  g1[5] = (int)ustr;                             // tensor_dim0_stride [31:0]
  g1[6] = 0;                                     // stride hi, dim1_stride lo
  g1[7] = 0;
  v4i z4 = {0, 0, 0, 0};
#if defined(__clang_major__) && (__clang_major__ >= 23)
  v8i z8 = {0, 0, 0, 0, 0, 0, 0, 0};
  __builtin_amdgcn_tensor_load_to_lds(g0, g1, z4, z4, z8, 0);
#else
  __builtin_amdgcn_tensor_load_to_lds(g0, g1, z4, z4, 0);
#endif
#else
  (void)gptr; (void)ldsAddr; (void)rowsAvail; (void)cols; (void)rowStride;
#endif
}

// ---------------------------------------------------------------------------
// gconv_gg as implicit GEMM.  Each wave32 computes a 16(Cout) x 16(L) tile
// for one (batch b, output-scale s); the (up to) 4 waves of a block share one
// TDM-staged LDS weight tile and cover 4 adjacent L-tiles.
// Loops: g (scale-group), kk (tap), then GEMM K = Cin stepped by 4 per
// V_WMMA_F32_16X16X4_F32; K loop split into unmasked body + masked tail so
// the steady state is 2 ds_load + 2 global_load + wmma with running pointers.
//
// A (16x4 fp32) VGPR layout (ISA 7.12.2): lanes 0-15 hold K={0,1} in v[0],
// v[1]; lanes 16-31 hold K={2,3}.  B (4x16) mirrored.  C/D 16x16: vgpr r ->
// row r (+8 for upper half-lanes), col = lane&15.
// ---------------------------------------------------------------------------
__global__ void gg_wmma_kernel(const float* __restrict__ x,
                               const float* __restrict__ w,
                               float* __restrict__ out,
                               int B, int Cin, int Sin, int L,
                               int Cout, int Ng) {
  extern __shared__ float smem[];   // [16][Ktot] weight tile

  const int lane   = threadIdx.x & 31;
  const int waveId = threadIdx.x >> 5;
  const int nWaves = blockDim.x >> 5;
  const int m      = lane & 15;     // row (A) / col (B) within tile
  const int khalf  = lane >> 4;     // 0: K={0,1}, 1: K={2,3}
  const int s      = blockIdx.y;    // output scale index
  const int coutTiles = (Cout + 15) >> 4;
  const int b   = blockIdx.z / coutTiles;
  const int co0 = (blockIdx.z % coutTiles) << 4;
  const int l0  = (blockIdx.x * nWaves + waveId) << 4;
  const int dil = 1 << s;
  const int Ng3 = Ng * 3;
  const int Ktot = Cin * Ng3;
  const int Sout = gridDim.y;
  const float scale = exp2f(-(float)s);      // the / 2^i reference factor

  // Wave 0 kicks off the TDM weight-tile load and waits; barrier releases the
  // sibling waves (TDM completion only signals the issuing wave).
  // LDS aperture mapping: flat LDS address == addr[31:0] (ISA 10.2).
  if (waveId == 0) {
    tdm_load_w_tile(w + (long long)co0 * Ktot,
                    (unsigned)(size_t)smem, Cout - co0, Ktot, Ktot);
#if __has_builtin(__builtin_amdgcn_s_wait_tensorcnt)
    __builtin_amdgcn_s_wait_tensorcnt(0);
#endif
#if !__has_builtin(__builtin_amdgcn_tensor_load_to_lds)
    for (int i = lane; i < 16 * Ktot; i += 32) {
      const int r = i / Ktot, c = i % Ktot;
      smem[i] = (co0 + r < Cout) ? w[(long long)(co0 + r) * Ktot + c] : 0.0f;
    }
#endif
  }
  __syncthreads();

  const int col = l0 + m;
  const int mKt = m * Ktot;
  const long long SinL = (long long)Sin * L;
  const float* xpl = x + ((long long)b * Cin * Sin) * L;   // batch plane

  const int nFull = Cin >> 2;       // unmasked 4-wide K steps
  const int rem   = Cin & 3;        // masked tail (45 -> 1, 90 -> 2)

  v8f acc = {};

  for (int g = 0; g < Ng; ++g) {
    const float* xg = xpl + (long long)(s + g) * L;        // (ci=0, s+g) row
    for (int kk = 0; kk < 3; ++kk) {
      const int wcol0 = g * 3 + kk;                        // K-col for ci=0
      const int lx = col + (kk - 1) * dil;                 // SAME pad, k=3
      const float bmask = (lx >= 0 && lx < L) ? 1.0f : 0.0f;
      const int lxc = lx < 0 ? 0 : (lx >= L ? L - 1 : lx);
      // Running pointers: ci0 = khalf*2, advancing by 4 per step.
      const float* xp = xg + lxc + (long long)(khalf * 2) * SinL;
      int ldsOff = mKt + (khalf * 2) * Ng3 + wcol0;
#pragma unroll 2
      for (int it = 0; it < nFull; ++it) {
        v2f a, bb;
        a.x  = smem[ldsOff];                   // ds_load, imm offsets
        a.y  = smem[ldsOff + Ng3];
        bb.x = xp[0]    * bmask;               // global_load, imm offsets
        bb.y = xp[SinL] * bmask;
        acc = __builtin_amdgcn_wmma_f32_16x16x4_f32(
            false, a, false, bb, (short)0, acc, false, false);
        xp     += 4 * SinL;
        ldsOff += 4 * Ng3;
      }
      if (rem) {                               // masked tail K step
        const int ci0 = nFull * 4 + khalf * 2;
        const int ci1 = ci0 + 1;
        const float am0 = (ci0 < Cin) ? 1.0f : 0.0f;
        const float am1 = (ci1 < Cin) ? 1.0f : 0.0f;
        const int c0 = (ci0 < Cin) ? ci0 : Cin - 1;
        const int c1 = (ci1 < Cin) ? ci1 : Cin - 1;
        v2f a, bb;
        a.x  = smem[mKt + c0 * Ng3 + wcol0] * am0;
        a.y  = smem[mKt + c1 * Ng3 + wcol0] * am1;
        bb.x = xg[lxc + (long long)c0 * SinL] * (am0 * bmask);
        bb.y = xg[lxc + (long long)c1 * SinL] * (am1 * bmask);
        acc = __builtin_amdgcn_wmma_f32_16x16x4_f32(
            false, a, false, bb, (short)0, acc, false, false);
      }
    }
  }

#pragma unroll
  for (int r = 0; r < 8; ++r) {
    const int row = co0 + r + khalf * 8;
    if (row < Cout)
      out[(((long long)b * Cout + row) * Sout + s) * L + col] = acc[r] * scale;
  }
}

// ---------------------------------------------------------------------------
// Lift conv (Cin=1, k=79, SAME, dilation 2^s, scaled by 2^-s) fused with the
// first pool4.  out shape (B, 45, 9, Lin/4).
// ---------------------------------------------------------------------------
__global__ void lift_pool_kernel(const float* __restrict__ x,
                                 const float* __restrict__ w,
                                 float* __restrict__ out,
                                 int B, int Lin, int Lout, int Cout, int NS) {
  const long long idx = (long long)blockIdx.x * blockDim.x + threadIdx.x;
  const long long total = (long long)B * Cout * NS * Lout;
  if (idx >= total) return;
  int lp = (int)(idx % Lout);
  long long t = idx / Lout;
  const int s = (int)(t % NS); t /= NS;
  const int c = (int)(t % Cout);
  const int b = (int)(t / Cout);
  const int dil = 1 << s;
  const float scale = exp2f(-(float)s);
  const float* wr = w + (long long)c * 79;
  const float* xr = x + (long long)b * Lin;
  float mx = -__builtin_inff();
  for (int p = 0; p < 4; ++p) {
    const int l = lp * 4 + p;
    float acc = 0.0f;
    for (int k = 0; k < 79; ++k) {
      const int lx = l + (k - 39) * dil;         // SAME pad for k=79
      if (lx >= 0 && lx < Lin) acc = fmaf(wr[k], xr[lx], acc);
    }
    mx = fmaxf(mx, acc * scale);
  }
  out[idx] = mx;
}

// ---------------------------------------------------------------------------
// BatchNorm statistics: one block per channel, reduce over (B, S, L).
// stats[2c] = mean, stats[2c+1] = rsqrt(var + eps).
// ---------------------------------------------------------------------------
__global__ void bn_stats_kernel(const float* __restrict__ x,
                                float* __restrict__ stats,
                                int nB, int SL, long long CSL) {
  const int c = blockIdx.x;
  __shared__ float s1[256];
  __shared__ float s2[256];
  float a1 = 0.0f, a2 = 0.0f;
  for (int b = 0; b < nB; ++b) {
    const float* p = x + (long long)b * CSL + (long long)c * SL;
    for (int i = threadIdx.x; i < SL; i += blockDim.x) {
      const float v = p[i];
      a1 += v; a2 += v * v;
    }
  }
  s1[threadIdx.x] = a1; s2[threadIdx.x] = a2;
  __syncthreads();
  for (int off = 128; off > 0; off >>= 1) {
    if ((int)threadIdx.x < off) {
      s1[threadIdx.x] += s1[threadIdx.x + off];
      s2[threadIdx.x] += s2[threadIdx.x + off];
    }
    __syncthreads();
  }
  if (threadIdx.x == 0) {
    const float n = (float)nB * (float)SL;
    const float mean = s1[0] / n;
    const float var  = s2[0] / n - mean * mean;
    stats[2 * c]     = mean;
    stats[2 * c + 1] = rsqrtf(var + BN_EPS);
  }
}

__global__ void bn_apply_kernel(float* __restrict__ y,
                                const float* __restrict__ stats,
                                long long total, int SL, int C, int relu) {
  const long long idx = (long long)blockIdx.x * blockDim.x + threadIdx.x;
  if (idx >= total) return;
  const int c = (int)((idx / SL) % C);
  float v = (y[idx] - stats[2 * c]) * stats[2 * c + 1];
  if (relu) v = fmaxf(v, 0.0f);
  y[idx] = v;
}

// ---------------------------------------------------------------------------
// Residual add.  Handles all three residual flavors by index mapping:
//   plain:     Cres==C, Sres==S     -> res[b,c,s,l]
//   override:  Cres==C, Sres==S+2   -> res[b,c,s<Sout,l] (slice)
//   diff:      Cres==C/2, Sres==S+2 -> res[b,c%Cres,s,l] (tile+slice)
// ---------------------------------------------------------------------------
__global__ void add_res_kernel(float* __restrict__ y,
                               const float* __restrict__ res,
                               int B, int C, int S, int L, int Cres, int Sres) {
  const long long idx = (long long)blockIdx.x * blockDim.x + threadIdx.x;
  const long long total = (long long)B * C * S * L;
  if (idx >= total) return;
  const int l = (int)(idx % L);
  long long t = idx / L;
  const int s = (int)(t % S); t /= S;
  const int c = (int)(t % C);
  const int b = (int)(t / C);
  y[idx] += res[(((long long)b * Cres + (c % Cres)) * Sres + s) * L + l];
}

__global__ void pool4_kernel(const float* __restrict__ x,
                             float* __restrict__ y,
                             long long total, int Lout) {
  const long long idx = (long long)blockIdx.x * blockDim.x + threadIdx.x;
  if (idx >= total) return;
  const int l = (int)(idx % Lout);
  const long long row = idx / Lout;
  const float* p = x + row * ((long long)Lout * 4) + (long long)l * 4;
  y[idx] = fmaxf(fmaxf(p[0], p[1]), fmaxf(p[2], p[3]));
}

__global__ void meanL_kernel(const float* __restrict__ x,
                             float* __restrict__ y,
                             long long total, int L) {
  const long long idx = (long long)blockIdx.x * blockDim.x + threadIdx.x;
  if (idx >= total) return;
  float a = 0.0f;
  const float* p = x + idx * L;
  for (int l = 0; l < L; ++l) a += p[l];
  y[idx] = a / (float)L;
}

// Final 1x1 gconv (Ng=1, k=1, dilation 1, /2^0) + max over singleton scale
// axis == plain channel projection: out[b,co] = sum_ci W[co,ci] * xm[b,ci].
__global__ void final_kernel(const float* __restrict__ xm,
                             const float* __restrict__ w,
                             float* __restrict__ out,
                             int B, int Cin, int Cout) {
  const int idx = blockIdx.x * blockDim.x + threadIdx.x;
  if (idx >= B * Cout) return;
  const int co = idx % Cout;
  const int b  = idx / Cout;
  float a = 0.0f;
  for (int ci = 0; ci < Cin; ++ci)
    a = fmaf(w[(long long)co * Cin + ci], xm[(long long)b * Cin + ci], a);
  out[idx] = a;
}

// ---------------------------------------------------------------------------
// Host-side orchestration
// ---------------------------------------------------------------------------
struct BlkCfg { int cin, cout, ng; };
static const BlkCfg kBlks[16] = {
  {45, 45, 3},  {45, 45, 1},  {45, 45, 1},
  {45, 90, 3},  {90, 90, 1},  {90, 90, 1},  {90, 90, 1},
  {90, 180, 3}, {180,180,1},  {180,180,1},  {180,180,1}, {180,180,1}, {180,180,1},
  {180,360, 3}, {360,360,1},  {360,360,1},
};
static const int kStageN[4] = {3, 4, 6, 3};

extern "C" void kernel_launch(void* const* d_in, const int* in_sizes, int n_in,
                              void* d_out, int out_size, void* d_ws, size_t ws_size,
                              hipStream_t stream) {
  (void)out_size; (void)ws_size;
  // Identify inputs by unique flat sizes; block weights keep their list order.
  const float* x = nullptr;
  const float* wlift = nullptr;
  const float* wout = nullptr;
  const float* wb[32];
  int nb = 0;
  for (int i = 0; i < n_in; ++i) {
    const int sz = in_sizes[i];
    if (sz == 8 * 16384 && !x)        x     = (const float*)d_in[i];
    else if (sz == 45 * 79 && !wlift) wlift = (const float*)d_in[i];
    else if (sz == 10 * 360 && !wout) wout  = (const float*)d_in[i];
    else if (nb < 32)                 wb[nb++] = (const float*)d_in[i];
  }

  // Allow dynamic LDS up to the largest weight tile: 16*1620*4 = 103680 B
  // (well inside the 320 KB WGP LDS budget on gfx1250).
  (void)hipFuncSetAttribute(reinterpret_cast<const void*>(gg_wmma_kernel),
                            hipFuncAttributeMaxDynamicSharedMemorySize, 116736);

  char* ws = (char*)d_ws;
  float* r0    = (float*)(ws);                          // 56 MB (lift tensor)
  float* r1    = (float*)(ws + ((size_t)56 << 20));     // 14 MB
  float* r2    = (float*)(ws + ((size_t)70 << 20));     // 14 MB
  float* stats = (float*)(ws + ((size_t)84 << 20));     // 2*360 floats

  const int B = 8, NS = 9;

  auto bn = [&](float* t, int C, int S, int L, int relu) {
    const int SL = S * L;
    const long long CSL = (long long)C * SL;
    bn_stats_kernel<<<C, 256, 0, stream>>>(t, stats, B, SL, CSL);
    const long long total = (long long)B * CSL;
    bn_apply_kernel<<<(unsigned)((total + 255) / 256), 256, 0, stream>>>(
        t, stats, total, SL, C, relu);
  };
  auto gg = [&](const float* in, const float* wp, float* outp,
                int Cin, int Sin, int L, int Cout, int Ng) {
    const int Sout = Sin - (Ng - 1);
    const int Ktot = Cin * Ng * 3;
    const int wpb  = (L >= 64) ? 4 : (L / 16);   // waves per block, exact tiling
    dim3 grid((unsigned)(L / (16 * wpb)), (unsigned)Sout,
              (unsigned)(B * ((Cout + 15) / 16)));
    gg_wmma_kernel<<<grid, 32 * wpb, (size_t)16 * Ktot * sizeof(float), stream>>>(
        in, wp, outp, B, Cin, Sin, L, Cout, Ng);
  };
  auto pool = [&](const float* in, float* outp, int C, int S, int L) {
    const long long total = (long long)B * C * S * (L / 4);
    pool4_kernel<<<(unsigned)((total + 255) / 256), 256, 0, stream>>>(
        in, outp, total, L / 4);
  };

  // 1) lift conv fused with first pool4: (8,45,9,4096) into r0
  {
    const long long total = (long long)B * 45 * NS * 4096;
    lift_pool_kernel<<<(unsigned)((total + 255) / 256), 256, 0, stream>>>(
        x, wlift, r0, B, 16384, 4096, 45, NS);
  }
  // 2) bn + relu, then second pool4 -> (8,45,9,1024) in r1
  bn(r0, 45, NS, 4096, 1);
  pool(r0, r1, 45, NS, 4096);

  float* cur = r1;  // current activation
  float* t1  = r2;
  float* t2  = r0;
  int S = NS, L = 1024, C = 45, bi = 0;

  for (int s = 0; s < 4; ++s) {
    for (int i = 0; i < kStageN[s]; ++i) {
      const BlkCfg c = kBlks[bi];
      const float* w1 = wb[2 * bi];
      const float* w2 = wb[2 * bi + 1];
      const int Sout = S - (c.ng - 1);

      gg(cur, w1, t1, c.cin, S, L, c.cout, c.ng);        // conv1
      bn(t1, c.cout, Sout, L, 1);                        // bn + relu
      gg(t1, w2, t2, c.cout, Sout, L, c.cout, 1);        // conv2
      bn(t2, c.cout, Sout, L, 0);                        // bn
      {                                                  // + residual
        const long long total = (long long)B * c.cout * Sout * L;
        add_res_kernel<<<(unsigned)((total + 255) / 256), 256, 0, stream>>>(
            t2, cur, B, c.cout, Sout, L, c.cin, S);
      }
      bn(t2, c.cout, Sout, L, 1);                        // bn + relu

      float* nx = t2; t2 = t1; t1 = cur; cur = nx;       // rotate buffers
      S = Sout; C = c.cout; ++bi;
    }
    if (s < 3) {
      pool(cur, t1, C, S, L);
      float* tmp = cur; cur = t1; t1 = tmp;
      L /= 4;
    }
  }

  // mean over L (shape is now (8,360,1,16)), then 360 -> 10 projection
  {
    const long long total = (long long)B * C;
    meanL_kernel<<<(unsigned)((total + 255) / 256), 256, 0, stream>>>(cur, t1, total, L);
  }
  final_kernel<<<1, 128, 0, stream>>>(t1, wout, (float*)d_out, B, C, 10);
}